// TGNN_Encoder_4380866642112
// MI455X (gfx1250) — compile-verified
//
#include <hip/hip_runtime.h>
#include <math.h>

// Problem constants from the reference
#define B_   1024
#define N_   200
#define DN_  128
#define DT_  128
#define M_   384      // DN + DN + DT

typedef __attribute__((ext_vector_type(2))) float v2f;
typedef __attribute__((ext_vector_type(8))) float v8f;

// ---------------------------------------------------------------------------
// CDNA5 async copy: global -> LDS, 128-bit per lane, tracked by ASYNCcnt.
// lds_addr is the low-32-bit LDS byte offset (truncated generic pointer).
// ---------------------------------------------------------------------------
__device__ __forceinline__ void async_load_b128(unsigned int lds_addr,
                                                const void* gaddr) {
    asm volatile("global_load_async_to_lds_b128 %0, %1, off"
                 :: "v"(lds_addr), "v"((unsigned long long)(uintptr_t)gaddr)
                 : "memory");
}
__device__ __forceinline__ void wait_async_le3() {
    asm volatile("s_wait_asynccnt 0x3" ::: "memory");
}
__device__ __forceinline__ void wait_async_0() {
    asm volatile("s_wait_asynccnt 0x0" ::: "memory");
}

// ---------------------------------------------------------------------------
// Kernel A: streaming attention with online softmax. HBM is read exactly once;
// keys stream via double-buffered GLOBAL_LOAD_ASYNC_TO_LDS_B128 (no VGPR
// round-trip), score-dot and weighted-sum both consume the LDS copy.
// grid = B blocks, 256 threads (8 wave32); wave w owns key row w of each chunk.
// ---------------------------------------------------------------------------
__global__ __launch_bounds__(256) void tgnn_attn_kernel(
    const float* __restrict__ src,      // [B,128]
    const float* __restrict__ src_t,    // [B,1,128]
    const float* __restrict__ seq,      // [B,200,128]
    const float* __restrict__ seq_t,    // [B,200,128]
    const float* __restrict__ seq_e,    // [B,200,128]
    const unsigned char* __restrict__ mask, // [B,200] bool
    const float* __restrict__ shared_attn,  // [1,768]
    float* __restrict__ h_out,          // ws: [B,384] attention output (pre-FC)
    float* __restrict__ attn_out)       // d_out + B*128 : [B,200]
{
    __shared__ float wk_s[M_];
    __shared__ float kbuf[2][8][M_];    // double-buffered key chunks (24 KB)
    __shared__ float scores_s[N_];
    __shared__ float red_s[256];

    const int b    = blockIdx.x;
    const int tid  = threadIdx.x;
    const int w    = tid >> 5;
    const int lane = tid & 31;

    // issue chunk 0 ASAP to overlap with the s_q reduction below
    {
        const long off = ((long)b * N_ + w) * 128 + 4*lane;
        float* row = &kbuf[0][w][0];
        async_load_b128((unsigned int)(uintptr_t)(row +       4*lane), seq   + off);
        async_load_b128((unsigned int)(uintptr_t)(row + 128 + 4*lane), seq_e + off);
        async_load_b128((unsigned int)(uintptr_t)(row + 256 + 4*lane), seq_t + off);
    }

    // preload wk = shared_attn[384:768] into LDS
    if (tid < 128) {
        wk_s[tid      ] = shared_attn[M_ + tid];
        wk_s[tid + 128] = shared_attn[M_ + 128 + tid];
        wk_s[tid + 256] = shared_attn[M_ + 256 + tid];
    }
    // s_q = dot(q, wq); q = [src, 0, src_t], wq = shared_attn[0:384]
    float part = 0.f;
    if (tid < 128) {
        part = src  [(long)b*DN_ + tid] * shared_attn[tid]
             + src_t[(long)b*DT_ + tid] * shared_attn[256 + tid];
    }
    red_s[tid] = part;
    __syncthreads();
    for (int off = 128; off > 0; off >>= 1) {
        if (tid < off) red_s[tid] += red_s[tid + off];
        __syncthreads();
    }
    const float s_q = red_s[0];

    const float4* wk4 = reinterpret_cast<const float4*>(wk_s);

    // online softmax state (uniform across the block)
    float mx = -3.0e38f, ssum = 0.f;
    float acc0 = 0.f, acc1 = 0.f;   // thread owns m=tid, and m=256+tid if tid<128

    for (int c = 0; c < 25; ++c) {          // 25 chunks * 8 rows = 200
        const int p = c & 1;
        const int r = c*8 + w;

        if (c + 1 < 25) {                   // async-stage next chunk into other buffer
            const long off = ((long)b * N_ + (r + 8)) * 128 + 4*lane;
            float* row = &kbuf[1 - p][w][0];
            async_load_b128((unsigned int)(uintptr_t)(row +       4*lane), seq   + off);
            async_load_b128((unsigned int)(uintptr_t)(row + 128 + 4*lane), seq_e + off);
            async_load_b128((unsigned int)(uintptr_t)(row + 256 + 4*lane), seq_t + off);
            wait_async_le3();               // in-order: chunk c now resident (own rows)
        } else {
            wait_async_0();
        }

        // score dot: wave reads its OWN row (loaded by its own async ops)
        const float4* kb4 = reinterpret_cast<const float4*>(&kbuf[p][w][0]);
        float4 a0 = kb4[lane], a1 = kb4[32 + lane], a2 = kb4[64 + lane];
        float4 w0 = wk4[lane], w1 = wk4[32 + lane], w2 = wk4[64 + lane];
        float d = a0.x*w0.x + a0.y*w0.y + a0.z*w0.z + a0.w*w0.w
                + a1.x*w1.x + a1.y*w1.y + a1.z*w1.z + a1.w*w1.w
                + a2.x*w2.x + a2.y*w2.y + a2.z*w2.z + a2.w*w2.w;
        #pragma unroll
        for (int o = 16; o > 0; o >>= 1) d += __shfl_xor(d, o, 32);

        if (lane == 0) {
            float sc = s_q + d;
            if (mask[(long)b*N_ + r]) sc = -1.0e10f;
            scores_s[r] = sc;
        }
        __syncthreads();   // all rows of chunk c + scores visible block-wide

        // uniform online-softmax update
        float cm = scores_s[c*8];
        #pragma unroll
        for (int j = 1; j < 8; ++j) cm = fmaxf(cm, scores_s[c*8 + j]);
        const float nm = fmaxf(mx, cm);
        const float f  = __expf(mx - nm);
        float pw[8], psum = 0.f;
        #pragma unroll
        for (int j = 0; j < 8; ++j) { pw[j] = __expf(scores_s[c*8 + j] - nm); psum += pw[j]; }
        ssum = ssum * f + psum;
        mx   = nm;

        float a0n = acc0 * f;
        #pragma unroll
        for (int j = 0; j < 8; ++j) a0n += pw[j] * kbuf[p][j][tid];
        acc0 = a0n;
        if (tid < 128) {
            float a1n = acc1 * f;
            #pragma unroll
            for (int j = 0; j < 8; ++j) a1n += pw[j] * kbuf[p][j][256 + tid];
            acc1 = a1n;
        }
        __syncthreads();   // buffer p free before next iteration's async writes
    }

    const float inv = 1.f / ssum;
    h_out[(long)b*M_ + tid] = acc0 * inv;
    if (tid < 128) h_out[(long)b*M_ + 256 + tid] = acc1 * inv;
    if (tid < N_)  attn_out[(long)b*N_ + tid] = __expf(scores_s[tid] - mx) * inv;
}

// ---------------------------------------------------------------------------
// Kernel B: fused dense tail with fp32 WMMA (V_WMMA_F32_16X16X4_F32).
// grid = B/16 blocks, 256 threads (8 waves). Each block owns 16 batch rows:
//   GEMM1 16x384 @ fc_w^T -> +q residual -> LayerNorm -> concat src (16x512)
//   GEMM2 @ w1^T + ReLU (16x256) -> GEMM3 @ w2^T -> out (16x128)
// LDS strides padded to 4 mod 64 to avoid bank conflicts.
// ---------------------------------------------------------------------------
#define HS_LD 388
#define OS_LD 516
#define XS_LD 260

__device__ __forceinline__ v8f wmma_f32(v8f c, const float* arow, const float* brow, int kk) {
    const float2 av = *reinterpret_cast<const float2*>(arow + kk);
    const float2 bv = *reinterpret_cast<const float2*>(brow + kk);
    v2f a; a.x = av.x; a.y = av.y;
    v2f b; b.x = bv.x; b.y = bv.y;
    return __builtin_amdgcn_wmma_f32_16x16x4_f32(false, a, false, b, (short)0, c, false, false);
}

__global__ __launch_bounds__(256) void tgnn_mlp_kernel(
    const float* __restrict__ h,      // [B,384] from kernel A
    const float* __restrict__ src,    // [B,128]
    const float* __restrict__ src_t,  // [B,128]
    const float* __restrict__ fc_w,   // [384,384]
    const float* __restrict__ ln_g,   // [384]
    const float* __restrict__ ln_b,   // [384]
    const float* __restrict__ w1,     // [256,512]
    const float* __restrict__ w2,     // [128,256]
    float* __restrict__ out)          // [B,128]
{
    __shared__ float Hs[16 * HS_LD];  // A for GEMM1
    __shared__ float Os[16 * OS_LD];  // GEMM1 out -> LN -> concat (16x512)
    __shared__ float Xs[16 * XS_LD];  // GEMM2 out (16x256)

    const int tid   = threadIdx.x;
    const int wv    = tid >> 5;
    const int lane  = tid & 31;
    const int ln16  = lane & 15;
    const int khalf = lane >> 4;         // A/B koff = 2*khalf
    const long b0   = (long)blockIdx.x * 16;

    // load H tile (16x384) with float4 global loads, padded LDS stores
    {
        const float4* g4 = reinterpret_cast<const float4*>(h + b0 * M_);
        #pragma unroll
        for (int i = 0; i < 6; ++i) {
            const int fi  = tid + 256*i;        // float4 index, 1536 total
            const float4 v = g4[fi];
            const int e   = fi * 4;
            const int row = e / M_;
            const int col = e - row * M_;
            float* d = &Hs[row * HS_LD + col];
            d[0] = v.x; d[1] = v.y; d[2] = v.z; d[3] = v.w;
        }
    }
    __syncthreads();

    // ---- GEMM1: C(16x384) = H(16x384) @ fc_w^T,  B[k][n] = fc_w[n*384+k]
    for (int nt = wv; nt < 24; nt += 8) {
        const int n0 = nt * 16;
        v8f c = {};
        const float* arow = Hs + ln16 * HS_LD;
        const float* brow = fc_w + (long)(n0 + ln16) * 384;
        #pragma unroll 4
        for (int k0 = 0; k0 < 384; k0 += 4)
            c = wmma_f32(c, arow, brow, k0 + 2*khalf);
        #pragma unroll
        for (int j = 0; j < 8; ++j) {
            const int m = j + 8*khalf;          // C layout: VGPR j, lanes>=16 -> M=j+8
            Os[m * OS_LD + n0 + ln16] = c[j];
        }
    }
    __syncthreads();

    // ---- residual + LayerNorm, then append src for concat. Wave handles 2 rows.
    #pragma unroll
    for (int rr = 0; rr < 2; ++rr) {
        const int  r = wv*2 + rr;
        const long bb = b0 + r;
        float sum = 0.f, sq = 0.f;
        float vb[12];
        #pragma unroll
        for (int i = 0; i < 12; ++i) {
            const int m = lane + 32*i;
            float qv = (m < 128) ? src[bb*128 + m]
                     : (m >= 256 ? src_t[bb*128 + (m - 256)] : 0.f);
            const float v = Os[r * OS_LD + m] + qv;
            vb[i] = v; sum += v; sq += v*v;
        }
        #pragma unroll
        for (int o = 16; o > 0; o >>= 1) {
            sum += __shfl_xor(sum, o, 32);
            sq  += __shfl_xor(sq,  o, 32);
        }
        const float mu  = sum * (1.f/384.f);
        const float var = sq * (1.f/384.f) - mu*mu;
        const float rs  = rsqrtf(var + 1e-5f);
        #pragma unroll
        for (int i = 0; i < 12; ++i) {
            const int m = lane + 32*i;
            Os[r * OS_LD + m] = (vb[i] - mu) * rs * ln_g[m] + ln_b[m];
        }
        #pragma unroll
        for (int i = 0; i < 4; ++i) {
            const int j = lane + 32*i;
            Os[r * OS_LD + 384 + j] = src[bb*128 + j];
        }
    }
    __syncthreads();

    // ---- GEMM2: X(16x256) = relu( Cat(16x512) @ w1^T ), B[k][n] = w1[n*512+k]
    for (int nt = wv; nt < 16; nt += 8) {
        const int n0 = nt * 16;
        v8f c = {};
        const float* arow = Os + ln16 * OS_LD;
        const float* brow = w1 + (long)(n0 + ln16) * 512;
        #pragma unroll 4
        for (int k0 = 0; k0 < 512; k0 += 4)
            c = wmma_f32(c, arow, brow, k0 + 2*khalf);
        #pragma unroll
        for (int j = 0; j < 8; ++j) {
            const int m = j + 8*khalf;
            Xs[m * XS_LD + n0 + ln16] = fmaxf(c[j], 0.f);
        }
    }
    __syncthreads();

    // ---- GEMM3: out(16x128) = X(16x256) @ w2^T, B[k][n] = w2[n*256+k]
    {
        const int n0 = wv * 16;
        v8f c = {};
        const float* arow = Xs + ln16 * XS_LD;
        const float* brow = w2 + (long)(n0 + ln16) * 256;
        #pragma unroll 4
        for (int k0 = 0; k0 < 256; k0 += 4)
            c = wmma_f32(c, arow, brow, k0 + 2*khalf);
        #pragma unroll
        for (int j = 0; j < 8; ++j) {
            const int m = j + 8*khalf;
            out[(b0 + m) * 128 + n0 + ln16] = c[j];
        }
    }
}

// ---------------------------------------------------------------------------
extern "C" void kernel_launch(void* const* d_in, const int* in_sizes, int n_in,
                              void* d_out, int out_size, void* d_ws, size_t ws_size,
                              hipStream_t stream) {
    // setup_inputs() order:
    const float* src         = (const float*)d_in[0];   // [1024,128]
    const float* src_t       = (const float*)d_in[1];   // [1024,1,128]
    const float* seq         = (const float*)d_in[2];   // [1024,200,128]
    const float* seq_t       = (const float*)d_in[3];   // [1024,200,128]
    const float* seq_e       = (const float*)d_in[4];   // [1024,200,128]
    const unsigned char* msk = (const unsigned char*)d_in[5]; // [1024,200] bool
    const float* shared_attn = (const float*)d_in[6];   // [1,768]
    const float* fc_w        = (const float*)d_in[7];   // [384,384]
    const float* ln_g        = (const float*)d_in[8];   // [384]
    const float* ln_b        = (const float*)d_in[9];   // [384]
    const float* agg_w1      = (const float*)d_in[10];  // [256,512]
    const float* agg_w2      = (const float*)d_in[11];  // [128,256]

    float* out_final = (float*)d_out;                   // [1024,128]
    float* out_attn  = (float*)d_out + (long)B_ * DN_;  // [1024,200]
    float* h_ws      = (float*)d_ws;                    // [1024,384] scratch

    tgnn_attn_kernel<<<B_, 256, 0, stream>>>(
        src, src_t, seq, seq_t, seq_e, msk, shared_attn, h_ws, out_attn);

    tgnn_mlp_kernel<<<B_ / 16, 256, 0, stream>>>(
        h_ws, src, src_t, fc_w, ln_g, ln_b, agg_w1, agg_w2, out_final);

    (void)in_sizes; (void)n_in; (void)out_size; (void)ws_size;
}